// HotspotOverlapLoss_45226005627632
// MI455X (gfx1250) — compile-verified
//
#include <hip/hip_runtime.h>

#define NB 8                      // batch
#define NPB (128*128*128)         // elements per batch = 2,097,152
#define TPB 256                   // threads per block
#define EPT 8                     // elements per thread (data passes)
#define BPB (NPB/(TPB*EPT))       // blocks per batch = 1024
#define GRID_DATA (NB*BPB)        // 8192

// -------- workspace layout (~1 MB) --------
struct Scratch {
    float sum_se[NB];
    float cnt_mask[NB];
    float cnt_pred[NB];
    float inter[NB];
    float phot[NB];
    float thot[NB];
    float th[2][NB];              // [0]=target threshold, [1]=pred threshold
    unsigned pref[2][2][NB];      // [arr][lo/hi][b] current bit-prefix (sentinel ~0u)
    unsigned rnk[2][2][NB];       // remaining rank within current prefix
    float    gfrac[2][NB];
    unsigned valid[2][NB];
    unsigned h0[2][NB][4096];     // level-0 hist: bits u>>19
    unsigned h1[2][2][NB][4096];  // level-1 hist: bits (u>>7)&0xFFF
    unsigned h2[2][2][NB][128];   // level-2 hist: bits u&0x7F
    float part1[3][NB][BPB];      // se, maskcnt, predcnt
    float part2[3][NB][BPB];      // inter, phot, thot
};

// -------- WMMA wave32 sum (exact f32, all lanes get total) --------
#if __has_builtin(__builtin_amdgcn_wmma_f32_16x16x4_f32)
typedef float v2f __attribute__((ext_vector_type(2)));
typedef float v8f __attribute__((ext_vector_type(8)));
__device__ __forceinline__ float wave_sum(float x) {
    // A(16x4): lane m holds K0(=x),K1(=0); lane m+16 holds K2(=x),K3(=0).
    // B(4x16)=all ones (layout-invariant), C=0 -> D[m][n]=x[m]+x[m+16].
    v2f a; a[0] = x; a[1] = 0.0f;
    v2f ones; ones[0] = 1.0f; ones[1] = 1.0f;
    v8f c = {};
    v8f d = __builtin_amdgcn_wmma_f32_16x16x4_f32(false, a, false, ones,
                                                  (short)0, c, false, false);
    // lanes<16: rows0..7 -> sum = sum_{m<8}(x[m]+x[m+16]); lanes>=16: rows8..15
    float s = d[0]+d[1]+d[2]+d[3]+d[4]+d[5]+d[6]+d[7];
    v2f a2; a2[0] = s; a2[1] = 0.0f;
    v8f d2 = __builtin_amdgcn_wmma_f32_16x16x4_f32(false, a2, false, ones,
                                                   (short)0, c, false, false);
    return d2[0];                 // s_lo + s_hi = grand total, every lane
}
#else
__device__ __forceinline__ float wave_sum(float x) {
    for (int o = 16; o > 0; o >>= 1) x += __shfl_down(x, o, 32);
    return x;                     // total in lane 0
}
#endif

__device__ __forceinline__ float block_sum(float x, float* red) {
    __syncthreads();
    float w = wave_sum(x);
    int lane = threadIdx.x & 31, wid = threadIdx.x >> 5;
    if (lane == 0) red[wid] = w;
    __syncthreads();
    if (threadIdx.x == 0) {
        float t = 0.f;
        for (int i = 0; i < TPB/32; ++i) t += red[i];   // fixed order
        red[0] = t;
    }
    __syncthreads();
    return red[0];
}

// -------- kernels --------
__global__ void hol_zero(unsigned* w) {
    const int words = (int)(sizeof(Scratch) / 4);
    for (int i = blockIdx.x*blockDim.x + threadIdx.x; i < words; i += gridDim.x*blockDim.x)
        w[i] = 0u;
}

__device__ __forceinline__ void pass1_elem(float t, float p, float& se, float& cm, float& cp,
                                           unsigned* sh_t, unsigned* sh_p) {
    if (t > 0.01f) {
        float d = p - t;
        se += d*d; cm += 1.f;
        atomicAdd(&sh_t[__float_as_uint(t) >> 19], 1u);
        if (p > 0.f) { cp += 1.f; atomicAdd(&sh_p[__float_as_uint(p) >> 19], 1u); }
    }
}

__global__ void hol_pass1(const float4* __restrict__ pred4, const float4* __restrict__ tgt4,
                          Scratch* ws) {
    __shared__ unsigned sh_t[4096];
    __shared__ unsigned sh_p[4096];
    __shared__ float red[TPB/32];
    int b = blockIdx.x >> 10, blk = blockIdx.x & (BPB-1);
    int base4 = (b*NPB + blk*(TPB*EPT)) >> 2;
    for (int i = threadIdx.x; i < 4096; i += TPB) { sh_t[i] = 0u; sh_p[i] = 0u; }
    __syncthreads();
    float se = 0.f, cm = 0.f, cp = 0.f;
    #pragma unroll
    for (int j = 0; j < EPT/4; ++j) {
        int idx4 = base4 + j*TPB + threadIdx.x;
        if (idx4 + TPB*(EPT/4) < (NB*NPB)/4) {
            __builtin_prefetch(&tgt4[idx4 + TPB*(EPT/4)], 0, 0);   // global_prefetch_b8
            __builtin_prefetch(&pred4[idx4 + TPB*(EPT/4)], 0, 0);
        }
        float4 t4 = tgt4[idx4], p4 = pred4[idx4];                  // global_load_b128
        pass1_elem(t4.x, p4.x, se, cm, cp, sh_t, sh_p);
        pass1_elem(t4.y, p4.y, se, cm, cp, sh_t, sh_p);
        pass1_elem(t4.z, p4.z, se, cm, cp, sh_t, sh_p);
        pass1_elem(t4.w, p4.w, se, cm, cp, sh_t, sh_p);
    }
    __syncthreads();
    for (int i = threadIdx.x; i < 4096; i += TPB) {
        unsigned c;
        c = sh_t[i]; if (c) atomicAdd(&ws->h0[0][b][i], c);
        c = sh_p[i]; if (c) atomicAdd(&ws->h0[1][b][i], c);
    }
    float s0 = block_sum(se, red);
    float s1 = block_sum(cm, red);
    float s2 = block_sum(cp, red);
    if (threadIdx.x == 0) {
        ws->part1[0][b][blk] = s0;
        ws->part1[1][b][blk] = s1;
        ws->part1[2][b][blk] = s2;
    }
}

__global__ void hol_reduce1(Scratch* ws) {
    __shared__ float red[TPB/32];
    int b = blockIdx.x;
    float s0 = 0.f, s1 = 0.f, s2 = 0.f;
    for (int i = threadIdx.x; i < BPB; i += TPB) {
        s0 += ws->part1[0][b][i]; s1 += ws->part1[1][b][i]; s2 += ws->part1[2][b][i];
    }
    s0 = block_sum(s0, red); s1 = block_sum(s1, red); s2 = block_sum(s2, red);
    if (threadIdx.x == 0) { ws->sum_se[b]=s0; ws->cnt_mask[b]=s1; ws->cnt_pred[b]=s2; }
}

// cooperative prefix over a histogram: returns total n, sh[tid]=exclusive chunk prefix
__device__ unsigned coop_prefix(const unsigned* hist, int nbins, unsigned* sh) {
    int per = (nbins + TPB - 1) / TPB;
    int lo = threadIdx.x * per;
    int hi = lo + per; if (hi > nbins) hi = nbins;
    unsigned s = 0;
    for (int i = lo; i < hi; ++i) s += hist[i];
    __syncthreads();
    sh[threadIdx.x] = s;
    __syncthreads();
    if (threadIdx.x == 0) {
        unsigned run = 0;
        for (int i = 0; i < TPB; ++i) { unsigned c = sh[i]; sh[i] = run; run += c; }
        sh[TPB] = run;
    }
    __syncthreads();
    return sh[TPB];
}

__device__ void coop_find(const unsigned* hist, int nbins, const unsigned* sh,
                          unsigned r, unsigned* outBin, unsigned* outRem) {
    int per = (nbins + TPB - 1) / TPB;
    int lo = threadIdx.x * per;
    int hi = lo + per; if (hi > nbins) hi = nbins;
    unsigned cum = sh[threadIdx.x];
    for (int i = lo; i < hi; ++i) {
        unsigned c = hist[i];
        if (r >= cum && r < cum + c) { *outBin = (unsigned)i; *outRem = r - cum; }
        cum += c;
    }
}

__global__ void hol_scan0(Scratch* ws) {
    __shared__ unsigned sh[TPB + 1];
    __shared__ unsigned res[4];
    __shared__ unsigned sRLo, sRHi, sValid;
    __shared__ float sG;
    int arr = blockIdx.x & 1, b = blockIdx.x >> 1;
    const unsigned* hist = ws->h0[arr][b];
    unsigned n = coop_prefix(hist, 4096, sh);
    if (threadIdx.x == 0) {
        if (n == 0) { sValid = 0u; }
        else {
            double pos = 0.9 * (double)(n - 1);
            unsigned i = (unsigned)pos;
            double g = pos - (double)i;
            if (i >= n - 1 && n > 1) { i = n - 1; g = 0.0; }
            if (n == 1) { i = 0; g = 0.0; }
            sRLo = i; sRHi = (i + 1 < n) ? (i + 1) : i;
            sG = (float)g; sValid = 1u;
        }
    }
    __syncthreads();
    if (sValid) {
        coop_find(hist, 4096, sh, sRLo, &res[0], &res[1]);
        coop_find(hist, 4096, sh, sRHi, &res[2], &res[3]);
        __syncthreads();
        if (threadIdx.x == 0) {
            ws->pref[arr][0][b] = res[0]; ws->rnk[arr][0][b] = res[1];
            ws->pref[arr][1][b] = res[2]; ws->rnk[arr][1][b] = res[3];
            ws->gfrac[arr][b] = sG; ws->valid[arr][b] = 1u;
        }
    } else if (threadIdx.x == 0) {
        ws->pref[arr][0][b] = 0xFFFFFFFFu; ws->pref[arr][1][b] = 0xFFFFFFFFu;
        ws->valid[arr][b] = 0u;
    }
}

__device__ __forceinline__ void ref1_elem(float t, float p, unsigned tl, unsigned th_,
                                          unsigned pl, unsigned ph_, Scratch* ws, int b) {
    if (t > 0.01f) {
        unsigned ut = __float_as_uint(t);
        unsigned hi12 = ut >> 19, mid12 = (ut >> 7) & 0xFFFu;
        if (hi12 == tl)  atomicAdd(&ws->h1[0][0][b][mid12], 1u);
        if (hi12 == th_) atomicAdd(&ws->h1[0][1][b][mid12], 1u);
        if (p > 0.f) {
            unsigned up = __float_as_uint(p);
            unsigned phi = up >> 19, pmid = (up >> 7) & 0xFFFu;
            if (phi == pl)  atomicAdd(&ws->h1[1][0][b][pmid], 1u);
            if (phi == ph_) atomicAdd(&ws->h1[1][1][b][pmid], 1u);
        }
    }
}

__global__ void hol_refine1(const float4* __restrict__ pred4, const float4* __restrict__ tgt4,
                            Scratch* ws) {
    int b = blockIdx.x >> 10, blk = blockIdx.x & (BPB-1);
    int base4 = (b*NPB + blk*(TPB*EPT)) >> 2;
    unsigned tl = ws->pref[0][0][b], th_ = ws->pref[0][1][b];
    unsigned pl = ws->pref[1][0][b], ph_ = ws->pref[1][1][b];
    #pragma unroll
    for (int j = 0; j < EPT/4; ++j) {
        int idx4 = base4 + j*TPB + threadIdx.x;
        float4 t4 = tgt4[idx4], p4 = pred4[idx4];
        ref1_elem(t4.x, p4.x, tl, th_, pl, ph_, ws, b);
        ref1_elem(t4.y, p4.y, tl, th_, pl, ph_, ws, b);
        ref1_elem(t4.z, p4.z, tl, th_, pl, ph_, ws, b);
        ref1_elem(t4.w, p4.w, tl, th_, pl, ph_, ws, b);
    }
}

__global__ void hol_scan1(Scratch* ws) {
    __shared__ unsigned sh[TPB + 1];
    __shared__ unsigned res[2];
    int arr = blockIdx.x & 1, b = blockIdx.x >> 1;
    if (!ws->valid[arr][b]) return;
    for (int sel = 0; sel < 2; ++sel) {
        const unsigned* hist = ws->h1[arr][sel][b];
        unsigned r = ws->rnk[arr][sel][b];
        coop_prefix(hist, 4096, sh);
        coop_find(hist, 4096, sh, r, &res[0], &res[1]);
        __syncthreads();
        if (threadIdx.x == 0) {
            ws->pref[arr][sel][b] = (ws->pref[arr][sel][b] << 12) | res[0];
            ws->rnk[arr][sel][b] = res[1];
        }
        __syncthreads();
    }
}

__device__ __forceinline__ void ref2_elem(float t, float p, unsigned tl, unsigned th_,
                                          unsigned pl, unsigned ph_, Scratch* ws, int b) {
    if (t > 0.01f) {
        unsigned ut = __float_as_uint(t);
        unsigned top24 = ut >> 7, low7 = ut & 0x7Fu;
        if (top24 == tl)  atomicAdd(&ws->h2[0][0][b][low7], 1u);
        if (top24 == th_) atomicAdd(&ws->h2[0][1][b][low7], 1u);
        if (p > 0.f) {
            unsigned up = __float_as_uint(p);
            unsigned ptop = up >> 7, plow = up & 0x7Fu;
            if (ptop == pl)  atomicAdd(&ws->h2[1][0][b][plow], 1u);
            if (ptop == ph_) atomicAdd(&ws->h2[1][1][b][plow], 1u);
        }
    }
}

__global__ void hol_refine2(const float4* __restrict__ pred4, const float4* __restrict__ tgt4,
                            Scratch* ws) {
    int b = blockIdx.x >> 10, blk = blockIdx.x & (BPB-1);
    int base4 = (b*NPB + blk*(TPB*EPT)) >> 2;
    unsigned tl = ws->pref[0][0][b], th_ = ws->pref[0][1][b];
    unsigned pl = ws->pref[1][0][b], ph_ = ws->pref[1][1][b];
    #pragma unroll
    for (int j = 0; j < EPT/4; ++j) {
        int idx4 = base4 + j*TPB + threadIdx.x;
        float4 t4 = tgt4[idx4], p4 = pred4[idx4];
        ref2_elem(t4.x, p4.x, tl, th_, pl, ph_, ws, b);
        ref2_elem(t4.y, p4.y, tl, th_, pl, ph_, ws, b);
        ref2_elem(t4.z, p4.z, tl, th_, pl, ph_, ws, b);
        ref2_elem(t4.w, p4.w, tl, th_, pl, ph_, ws, b);
    }
}

__global__ void hol_scan2(Scratch* ws) {
    __shared__ unsigned sh[TPB + 1];
    __shared__ unsigned res[2];
    __shared__ float vals[2];
    int arr = blockIdx.x & 1, b = blockIdx.x >> 1;
    if (!ws->valid[arr][b]) {
        if (threadIdx.x == 0) ws->th[arr][b] = 0.01f;
        return;
    }
    for (int sel = 0; sel < 2; ++sel) {
        const unsigned* hist = ws->h2[arr][sel][b];
        unsigned r = ws->rnk[arr][sel][b];
        coop_prefix(hist, 128, sh);
        coop_find(hist, 128, sh, r, &res[0], &res[1]);
        __syncthreads();
        if (threadIdx.x == 0)
            vals[sel] = __uint_as_float((ws->pref[arr][sel][b] << 7) | res[0]);
        __syncthreads();
    }
    if (threadIdx.x == 0) {
        float g = ws->gfrac[arr][b];
        ws->th[arr][b] = vals[0] + g * (vals[1] - vals[0]);
    }
}

__device__ __forceinline__ void fin_elem(float t, float p, float tht, float thp,
                                         float& ci, float& cp, float& ct) {
    if (t > 0.01f) {
        bool h_t = t > tht;     // masked_target > th (zeros never exceed th>0)
        bool h_p = p > thp;     // masked_pred   > th
        ct += h_t ? 1.f : 0.f;
        cp += h_p ? 1.f : 0.f;
        ci += (h_t && h_p) ? 1.f : 0.f;
    }
}

__global__ void hol_final(const float4* __restrict__ pred4, const float4* __restrict__ tgt4,
                          Scratch* ws) {
    __shared__ float red[TPB/32];
    int b = blockIdx.x >> 10, blk = blockIdx.x & (BPB-1);
    int base4 = (b*NPB + blk*(TPB*EPT)) >> 2;
    float tht = ws->th[0][b], thp = ws->th[1][b];
    float ci = 0.f, cp = 0.f, ct = 0.f;
    #pragma unroll
    for (int j = 0; j < EPT/4; ++j) {
        int idx4 = base4 + j*TPB + threadIdx.x;
        if (idx4 + TPB*(EPT/4) < (NB*NPB)/4) {
            __builtin_prefetch(&tgt4[idx4 + TPB*(EPT/4)], 0, 0);
            __builtin_prefetch(&pred4[idx4 + TPB*(EPT/4)], 0, 0);
        }
        float4 t4 = tgt4[idx4], p4 = pred4[idx4];
        fin_elem(t4.x, p4.x, tht, thp, ci, cp, ct);
        fin_elem(t4.y, p4.y, tht, thp, ci, cp, ct);
        fin_elem(t4.z, p4.z, tht, thp, ci, cp, ct);
        fin_elem(t4.w, p4.w, tht, thp, ci, cp, ct);
    }
    float s0 = block_sum(ci, red);
    float s1 = block_sum(cp, red);
    float s2 = block_sum(ct, red);
    if (threadIdx.x == 0) {
        ws->part2[0][b][blk] = s0;
        ws->part2[1][b][blk] = s1;
        ws->part2[2][b][blk] = s2;
    }
}

__global__ void hol_reduce2(Scratch* ws) {
    __shared__ float red[TPB/32];
    int b = blockIdx.x;
    float s0 = 0.f, s1 = 0.f, s2 = 0.f;
    for (int i = threadIdx.x; i < BPB; i += TPB) {
        s0 += ws->part2[0][b][i]; s1 += ws->part2[1][b][i]; s2 += ws->part2[2][b][i];
    }
    s0 = block_sum(s0, red); s1 = block_sum(s1, red); s2 = block_sum(s2, red);
    if (threadIdx.x == 0) { ws->inter[b]=s0; ws->phot[b]=s1; ws->thot[b]=s2; }
}

__global__ void hol_finish(Scratch* ws, float* out) {
    if (threadIdx.x == 0 && blockIdx.x == 0) {
        float acc = 0.f;
        for (int b = 0; b < NB; ++b) {
            float mse  = ws->sum_se[b] / (ws->cnt_mask[b] + 1e-8f);
            float dice = 1.f - 2.f * ws->inter[b] / (ws->phot[b] + ws->thot[b] + 1e-8f);
            acc += 0.5f * mse + 0.5f * dice;
        }
        out[0] = acc / (float)NB;
    }
}

extern "C" void kernel_launch(void* const* d_in, const int* in_sizes, int n_in,
                              void* d_out, int out_size, void* d_ws, size_t ws_size,
                              hipStream_t stream) {
    const float4* pred4 = (const float4*)d_in[0];
    const float4* tgt4  = (const float4*)d_in[1];
    float* out = (float*)d_out;
    Scratch* ws = (Scratch*)d_ws;

    hol_zero   <<<dim3(2048),      dim3(TPB), 0, stream>>>((unsigned*)d_ws);
    hol_pass1  <<<dim3(GRID_DATA), dim3(TPB), 0, stream>>>(pred4, tgt4, ws);
    hol_reduce1<<<dim3(NB),        dim3(TPB), 0, stream>>>(ws);
    hol_scan0  <<<dim3(16),        dim3(TPB), 0, stream>>>(ws);
    hol_refine1<<<dim3(GRID_DATA), dim3(TPB), 0, stream>>>(pred4, tgt4, ws);
    hol_scan1  <<<dim3(16),        dim3(TPB), 0, stream>>>(ws);
    hol_refine2<<<dim3(GRID_DATA), dim3(TPB), 0, stream>>>(pred4, tgt4, ws);
    hol_scan2  <<<dim3(16),        dim3(TPB), 0, stream>>>(ws);
    hol_final  <<<dim3(GRID_DATA), dim3(TPB), 0, stream>>>(pred4, tgt4, ws);
    hol_reduce2<<<dim3(NB),        dim3(TPB), 0, stream>>>(ws);
    hol_finish <<<dim3(1),         dim3(32),  0, stream>>>(ws, out);
}